// HMM_4578435137592
// MI455X (gfx1250) — compile-verified
//
#include <hip/hip_runtime.h>
#include <math.h>

// CDNA5 f32 WMMA fragments: A 16x4 / B 4x16 = 2 VGPRs, C/D 16x16 = 8 VGPRs (wave32)
typedef __attribute__((ext_vector_type(2))) float v2f;
typedef __attribute__((ext_vector_type(8))) float v8f;

#define NT    256          // 8 wave32s, one WGP, whole filter lives in LDS
#define NW    8
#define TLEN  4096
#define OBS   64           // obs_dim = 32 sensors * 2
#define DD    66           // augmented state dim
#define NRHS  67           // 66 cols of PHt^T + 1 col for r
#define PLD   84           // strides chosen for conflict-free column reads (mod 64)
#define LLD   68
#define YLD   84
#define LOG2PIF 1.8378770664093453f

// dynamic-LDS partition (floats)
#define SP_OFF 0
#define SL_OFF (80 * PLD)              // sP: 80x84 (66x66 live, rest write-only pad)
#define SY_OFF (SL_OFF + 64 * LLD)     // sL: 64x68
#define SM_OFF (SY_OFF + 68 * YLD)     // sY: 68x84 (cols >=67 stay exactly 0)
#define ID_OFF (SM_OFF + 68)
#define SA_OFF (ID_OFF + 68)
#define SC_OFF (SA_OFF + 4)
#define SMEM_FLOATS (SC_OFF + 8)
#define SMEM_BYTES  (SMEM_FLOATS * 4)

__device__ __forceinline__ v8f wmma4(v2f a, v2f b, v8f c) {
  return __builtin_amdgcn_wmma_f32_16x16x4_f32(false, a, false, b, (short)0, c, false, false);
}

__global__ __launch_bounds__(NT, 1)
void kf_hmm_kernel(const float* __restrict__ trk,
                   const float* __restrict__ bias_scales,
                   const float* __restrict__ obs_noise,
                   const float* __restrict__ trans_noise,
                   const float* __restrict__ Am,
                   const float* __restrict__ init_cov,
                   float* __restrict__ out)
{
  extern __shared__ float smem[];
  float* sP   = smem + SP_OFF;   // covariance P
  float* sL   = smem + SL_OFF;   // Sy -> Cholesky factor L
  float* sY   = smem + SY_OFF;   // RHS / Y = L^{-1}[PHt^T | r]; Z at the end
  float* sm   = smem + SM_OFF;   // mean
  float* invd = smem + ID_OFF;   // 1/diag(L)
  float* sA   = smem + SA_OFF;   // transition matrix
  float* sc   = smem + SC_OFF;   // 0=ll 1=logdet 2,3=1/d (double buffer) 4=r2 5=q2

  const int tid  = threadIdx.x;
  const int lane = tid & 31;
  const int wid  = tid >> 5;
  const int lo   = lane & 15;
  const int hi   = lane >> 4;

  // ---------------- init ----------------
  for (int idx = tid; idx < SMEM_FLOATS; idx += NT) smem[idx] = 0.0f;
  __syncthreads();
  if (tid == 0) {
    sA[0] = Am[0]; sA[1] = Am[1]; sA[2] = Am[2]; sA[3] = Am[3];
    float on = obs_noise[0], tn = trans_noise[0];
    sc[4] = on * on;
    sc[5] = tn * tn;
    sP[64 * PLD + 64] = init_cov[0]; sP[64 * PLD + 65] = init_cov[1];
    sP[65 * PLD + 64] = init_cov[2]; sP[65 * PLD + 65] = init_cov[3];
  }
  if (tid < OBS) sP[tid * PLD + tid] = bias_scales[tid & 1];  // tile(bias_scales, S)
  __syncthreads();

  // ---------------- scan ----------------
  for (int t = 0; t < TLEN; ++t) {
    if (tid == 0 && t + 1 < TLEN) __builtin_prefetch(&trk[2 * (t + 1)], 0, 3);

    // (1) predict: F = blockdiag(I64, A^T)
    float pv = 0.f, t00 = 0.f, t01 = 0.f, t10 = 0.f, t11 = 0.f, nm0 = 0.f, nm1 = 0.f;
    if (tid < 128) {                       // P_bs' = P_bs @ A
      int i = tid >> 1, s = tid & 1;
      float p0 = sP[i * PLD + 64], p1 = sP[i * PLD + 65];
      pv = p0 * sA[s] + p1 * sA[2 + s];
    } else if (tid == 128) {               // P_ss' = A^T P_ss A + q2 I
      float A00 = sA[0], A01 = sA[1], A10 = sA[2], A11 = sA[3];
      float p00 = sP[64*PLD+64], p01 = sP[64*PLD+65], p10 = sP[65*PLD+64], p11 = sP[65*PLD+65];
      float b00 = p00*A00 + p01*A10, b01 = p00*A01 + p01*A11;
      float b10 = p10*A00 + p11*A10, b11 = p10*A01 + p11*A11;
      t00 = A00*b00 + A10*b10; t01 = A00*b01 + A10*b11;
      t10 = A01*b00 + A11*b10; t11 = A01*b01 + A11*b11;
    } else if (tid == 129) {               // m_s' = A^T m_s
      nm0 = sA[0]*sm[64] + sA[2]*sm[65];
      nm1 = sA[1]*sm[64] + sA[3]*sm[65];
    }
    __syncthreads();
    if (tid < 128) {
      int i = tid >> 1, s = tid & 1;
      sP[i * PLD + 64 + s] = pv;
      sP[(64 + s) * PLD + i] = pv;
    } else if (tid == 128) {
      float q2 = sc[5];
      sP[64*PLD+64] = t00 + q2; sP[64*PLD+65] = t01;
      sP[65*PLD+64] = t10;      sP[65*PLD+65] = t11 + q2;
    } else if (tid == 129) { sm[64] = nm0; sm[65] = nm1; }
    __syncthreads();

    // (2) fused: Sy build + RHS build + first pivot
    {
      float r2 = sc[4];
      for (int idx = tid; idx < OBS * OBS; idx += NT) {
        int i = idx >> 6, j = idx & 63;
        int gi = i >> 5, gj = j >> 5;
        float v = sP[i*PLD + j] + sP[(64+gi)*PLD + j]
                + sP[i*PLD + 64 + gj] + sP[(64+gi)*PLD + 64 + gj];
        if (i == j) v += r2;
        sL[i * LLD + j] = v;
      }
      float x0 = trk[2 * t], x1 = trk[2 * t + 1];
      for (int idx = tid; idx < NRHS * OBS; idx += NT) {   // idx = c*64 + k (div-free)
        int c = idx >> 6, k = idx & 63;
        float v;
        if (c < DD) v = sP[c * PLD + k] + sP[c * PLD + 64 + (k >> 5)];
        else        v = ((k & 1) ? x1 : x0) - sm[k] - sm[64 + (k >> 5)];
        sY[k * YLD + c] = v;
      }
      if (tid == 0) {                      // pivot 0 (recompute Sy(0,0) locally)
        float d = sP[0] + sP[64*PLD] + sP[64] + sP[64*PLD+64] + r2;
        float s = sqrtf(d);
        sL[0] = s; invd[0] = 1.0f / s;
        sc[2] = 1.0f / d;
        sc[1] = logf(s);
      }
    }
    __syncthreads();

    // (3) Cholesky, single barrier per column: tid0 owns a(k+1,k+1)'s last update,
    //     so it also produces the next pivot in the same phase (1/d double-buffered).
    for (int k = 0; k < OBS - 1; ++k) {
      float rd = sc[2 + (k & 1)];
      for (int i = k + 1 + wid; i < OBS; i += NW) {
        float lik = sL[i * LLD + k] * rd;
        for (int j = k + 1 + lane; j <= i; j += 32)
          sL[i * LLD + j] -= lik * sL[j * LLD + k];
      }
      if (tid == 0) {
        float d = sL[(k + 1) * LLD + (k + 1)];
        float s = sqrtf(d);
        sL[(k + 1) * LLD + (k + 1)] = s;
        invd[k + 1] = 1.0f / s;
        sc[2 + ((k + 1) & 1)] = 1.0f / d;
        sc[1] += logf(s);
      }
      __syncthreads();
    }
    for (int idx = tid; idx < OBS * OBS; idx += NT) {  // deferred column scaling
      int i = idx >> 6, j = idx & 63;
      if (j < i) sL[i * LLD + j] *= invd[j];
    }
    __syncthreads();

    // (4) blocked forward TRSM: Y = L^{-1} [PHt^T | r]; WMMA rank-16 updates
    for (int kb = 0; kb < 4; ++kb) {
      if (tid < NRHS) {                 // 16x16 diagonal block, barrier-free per column
        int c = tid;
        #pragma unroll
        for (int ii = 0; ii < 16; ++ii) {
          int row = kb * 16 + ii;
          float acc = sY[row * YLD + c];
          for (int jj = 0; jj < ii; ++jj)
            acc -= sL[row * LLD + kb * 16 + jj] * sY[(kb * 16 + jj) * YLD + c];
          sY[row * YLD + c] = acc * invd[row];
        }
      }
      __syncthreads();
      if (kb < 3) {
        int ntiles = (3 - kb) * 5;      // B[jb] -= L[jb,kb] @ Y[kb]
        for (int tile = wid; tile < ntiles; tile += NW) {
          int jb = kb + 1 + tile / 5;
          int tj = tile - (tile / 5) * 5;
          int colc = tj * 16 + lo;
          const float* Lr = sL + (jb * 16 + lo) * LLD + kb * 16;
          const float* Yc = sY + kb * 16 * YLD + colc;
          v8f acc0, acc1;
          #pragma unroll
          for (int v = 0; v < 8; ++v) {
            acc0[v] = sY[(jb * 16 + hi * 8 + v) * YLD + colc];
            acc1[v] = 0.0f;
          }
          #pragma unroll
          for (int kc = 0; kc < 4; ++kc) {   // two independent WMMA chains
            int kk = kc * 4 + hi * 2;
            v2f a, b;
            a[0] = -Lr[kk];
            a[1] = -Lr[kk + 1];
            b[0] = Yc[kk * YLD];
            b[1] = Yc[(kk + 1) * YLD];
            if (kc & 1) acc1 = wmma4(a, b, acc1);
            else        acc0 = wmma4(a, b, acc0);
          }
          #pragma unroll
          for (int v = 0; v < 8; ++v)
            sY[(jb * 16 + hi * 8 + v) * YLD + colc] = acc0[v] + acc1[v];
        }
        __syncthreads();
      }
    }

    // (5) fused: log-lik (r^T Sy^-1 r = w.w), mean m += Y^T w, SYRK P -= Y^T Y
    if (tid < DD) {
      float acc = 0.0f;
      for (int k = 0; k < OBS; ++k)
        acc += sY[k * YLD + tid] * sY[k * YLD + DD];
      sm[tid] += acc;
    }
    if (tid == 0) {
      float ss = 0.0f;
      for (int k = 0; k < OBS; ++k) { float w = sY[k * YLD + DD]; ss += w * w; }
      sc[0] += -0.5f * (OBS * LOG2PIF + ss) - sc[1];
    }
    for (int tile = wid; tile < 15; tile += NW) {   // lower tiles of 5x5
      int ti = 0;
      while ((ti + 1) * (ti + 2) / 2 <= tile) ++ti;
      int tj = tile - ti * (ti + 1) / 2;
      int colA = ti * 16 + lo;
      int colB = tj * 16 + lo;
      v8f acc0, acc1;
      #pragma unroll
      for (int v = 0; v < 8; ++v) {
        acc0[v] = sP[(ti * 16 + hi * 8 + v) * PLD + colB];
        acc1[v] = 0.0f;
      }
      #pragma unroll
      for (int kc = 0; kc < 16; ++kc) {             // two independent WMMA chains
        int kk = kc * 4 + hi * 2;
        v2f a, b;
        a[0] = -sY[kk * YLD + colA];
        a[1] = -sY[(kk + 1) * YLD + colA];
        b[0] = sY[kk * YLD + colB];
        b[1] = sY[(kk + 1) * YLD + colB];
        if (kc & 1) acc1 = wmma4(a, b, acc1);
        else        acc0 = wmma4(a, b, acc0);
      }
      #pragma unroll
      for (int v = 0; v < 8; ++v)
        sP[(ti * 16 + hi * 8 + v) * PLD + colB] = acc0[v] + acc1[v];
    }
    __syncthreads();

    // (6) symmetrize (mirror lower -> upper)
    for (int idx = tid; idx < DD * DD; idx += NT) {
      int i = idx / DD, j = idx - i * DD;
      if (i < j) sP[i * PLD + j] = sP[j * PLD + i];
    }
    __syncthreads();
  }

  // ---------------- precision = inv(P_T) via Cholesky ----------------
  if (tid == 0) {
    float d = sP[0];
    float s = sqrtf(d);
    sP[0] = s; invd[0] = 1.0f / s; sc[2] = 1.0f / d;
  }
  __syncthreads();
  for (int k = 0; k < DD - 1; ++k) {
    float rd = sc[2 + (k & 1)];
    for (int i = k + 1 + wid; i < DD; i += NW) {
      float lik = sP[i * PLD + k] * rd;
      for (int j = k + 1 + lane; j <= i; j += 32)
        sP[i * PLD + j] -= lik * sP[j * PLD + k];
    }
    if (tid == 0) {
      float d = sP[(k + 1) * PLD + (k + 1)];
      float s = sqrtf(d);
      sP[(k + 1) * PLD + (k + 1)] = s;
      invd[k + 1] = 1.0f / s;
      sc[2 + ((k + 1) & 1)] = 1.0f / d;
    }
    __syncthreads();
  }
  for (int idx = tid; idx < DD * DD; idx += NT) {
    int i = idx / DD, j = idx - i * DD;
    if (j < i) sP[i * PLD + j] *= invd[j];
  }
  __syncthreads();

  if (tid < DD) {                      // Z = L^{-1}, barrier-free per column
    int c = tid;
    for (int i = 0; i < DD; ++i) {
      float acc = (i == c) ? 1.0f : 0.0f;
      for (int j = 0; j < i; ++j)
        acc -= sP[i * PLD + j] * sY[j * YLD + c];
      sY[i * YLD + c] = acc * invd[i];
    }
  }
  __syncthreads();

  for (int idx = tid; idx < DD * DD; idx += NT) {   // precision = Z^T Z
    int i = idx / DD, j = idx - i * DD;
    float acc = 0.0f;
    for (int k = 0; k < DD; ++k)
      acc += sY[k * YLD + i] * sY[k * YLD + j];
    out[1 + idx] = acc;
  }
  if (tid == 0) out[0] = sc[0];
}

extern "C" void kernel_launch(void* const* d_in, const int* in_sizes, int n_in,
                              void* d_out, int out_size, void* d_ws, size_t ws_size,
                              hipStream_t stream) {
  (void)in_sizes; (void)n_in; (void)d_ws; (void)ws_size; (void)out_size;
  const float* trk = (const float*)d_in[0];
  const float* bs  = (const float*)d_in[1];
  const float* on  = (const float*)d_in[2];
  const float* tn  = (const float*)d_in[3];
  const float* Am  = (const float*)d_in[4];
  const float* ic  = (const float*)d_in[5];
  float* out = (float*)d_out;
  kf_hmm_kernel<<<1, NT, SMEM_BYTES, stream>>>(trk, bs, on, tn, Am, ic, out);
}